// EdgeTrans_85074712199736
// MI455X (gfx1250) — compile-verified
//
#include <hip/hip_runtime.h>
#include <math.h>

// Problem constants (from reference)
#define NN 50000
#define EE 800000
#define HH 4
#define LDS_ST 68   // 16-row tile stride in floats (padded: 68*4=272B, 16B aligned, conflict-free)

typedef __attribute__((ext_vector_type(2))) float v2f;
typedef __attribute__((ext_vector_type(8))) float v8f;

// fp32 WMMA: D(16x16,f32) = A(16x4,f32) x B(4x16,f32) + C
__device__ __forceinline__ v8f wmma4(v2f a, v2f b, v8f c) {
  return __builtin_amdgcn_wmma_f32_16x16x4_f32(false, a, false, b, (short)0, c,
                                               false, false);
}

// order-preserving float<->uint encoding for atomicMax-based segment max
__device__ __forceinline__ unsigned encf(float f) {
  unsigned u = __float_as_uint(f);
  return (u & 0x80000000u) ? ~u : (u | 0x80000000u);
}
__device__ __forceinline__ float decf(unsigned u) {
  unsigned v = (u & 0x80000000u) ? (u ^ 0x80000000u) : ~u;
  return __uint_as_float(v);
}

// ---------------------------------------------------------------- init
__global__ void k_init(float* __restrict__ out_acc, unsigned* __restrict__ amax,
                       float* __restrict__ denom) {
  int i = blockIdx.x * blockDim.x + threadIdx.x;
  if (i < NN * 64) out_acc[i] = 0.f;
  if (i < NN * HH) {
    amax[i] = encf(-INFINITY);
    denom[i] = 0.f;
  }
}

// ------------------------------------------------- node GEMM: out = X@W + b
// one wave per 16-row tile; 4 column tiles of 16; K loop 64/4 = 16 -> 64 wmma
__global__ void __launch_bounds__(32) k_nodegemm(const float* __restrict__ X,
                                                 const float* __restrict__ W,
                                                 const float* __restrict__ bias,
                                                 float* __restrict__ out) {
  int tile = blockIdx.x;
  int lane = threadIdx.x;
  int n = lane & 15;
  int koff = (lane < 16) ? 0 : 2;
  int arow = tile * 16 + n;

  v8f c[4] = {};
  for (int kk = 0; kk < 16; ++kk) {
    int k0 = kk * 4 + koff;
    v2f a;
    a.x = X[(size_t)arow * 64 + k0];
    a.y = X[(size_t)arow * 64 + k0 + 1];
#pragma unroll
    for (int t = 0; t < 4; ++t) {
      int col = t * 16 + n;
      v2f b;
      b.x = W[(size_t)k0 * 64 + col];
      b.y = W[(size_t)(k0 + 1) * 64 + col];
      c[t] = wmma4(a, b, c[t]);
    }
  }
  int rbase = (lane < 16) ? 0 : 8;
#pragma unroll
  for (int t = 0; t < 4; ++t) {
    int col = t * 16 + n;
    float bb = bias[col];
#pragma unroll
    for (int r = 0; r < 8; ++r) {
      int row = tile * 16 + rbase + r;
      out[(size_t)row * 64 + col] = c[t][r] + bb;
    }
  }
}

// ---------------- per 16-edge tile: e = LeakyReLU(attr@We1+b1)@We2+b2 @ We
// result e (16x64) left in this wave's LDS slice
__device__ void edge_tile_e(int e0, int lane, float* lds,
                            const float* __restrict__ attr,
                            const float* __restrict__ We1,
                            const float* __restrict__ be1,
                            const float* __restrict__ We2,
                            const float* __restrict__ be2,
                            const float* __restrict__ Wed) {
  int n = lane & 15;
  int koff = (lane < 16) ? 0 : 2;
  int rbase = (lane < 16) ? 0 : 8;

  // GEMM1: h = attr(16x7, zero-padded to K=8) @ We1(7x64)
  v8f c[4] = {};
#pragma unroll
  for (int kk = 0; kk < 2; ++kk) {
    int k0 = kk * 4 + koff;
    v2f a;
    a.x = (k0 < 7) ? attr[(size_t)(e0 + n) * 7 + k0] : 0.f;
    a.y = (k0 + 1 < 7) ? attr[(size_t)(e0 + n) * 7 + k0 + 1] : 0.f;
#pragma unroll
    for (int t = 0; t < 4; ++t) {
      int col = t * 16 + n;
      v2f b;
      b.x = (k0 < 7) ? We1[(size_t)k0 * 64 + col] : 0.f;
      b.y = (k0 + 1 < 7) ? We1[(size_t)(k0 + 1) * 64 + col] : 0.f;
      c[t] = wmma4(a, b, c[t]);
    }
  }
#pragma unroll
  for (int t = 0; t < 4; ++t) {
    float bb = be1[t * 16 + n];
#pragma unroll
    for (int r = 0; r < 8; ++r) {
      float h = c[t][r] + bb;
      h = (h >= 0.f) ? h : 0.15f * h;  // LeakyReLU(0.15)
      lds[(rbase + r) * LDS_ST + t * 16 + n] = h;
    }
  }

  // GEMM2: emb = h @ We2 + b2   (A operand re-read from LDS in A-layout)
  v8f d[4] = {};
  for (int kk = 0; kk < 16; ++kk) {
    int k0 = kk * 4 + koff;
    v2f a;
    a.x = lds[n * LDS_ST + k0];
    a.y = lds[n * LDS_ST + k0 + 1];
#pragma unroll
    for (int t = 0; t < 4; ++t) {
      int col = t * 16 + n;
      v2f b;
      b.x = We2[(size_t)k0 * 64 + col];
      b.y = We2[(size_t)(k0 + 1) * 64 + col];
      d[t] = wmma4(a, b, d[t]);
    }
  }
#pragma unroll
  for (int t = 0; t < 4; ++t) {
    float bb = be2[t * 16 + n];
#pragma unroll
    for (int r = 0; r < 8; ++r)
      lds[(rbase + r) * LDS_ST + t * 16 + n] = d[t][r] + bb;
  }

  // GEMM3: e = emb @ We (no bias)
  v8f g[4] = {};
  for (int kk = 0; kk < 16; ++kk) {
    int k0 = kk * 4 + koff;
    v2f a;
    a.x = lds[n * LDS_ST + k0];
    a.y = lds[n * LDS_ST + k0 + 1];
#pragma unroll
    for (int t = 0; t < 4; ++t) {
      int col = t * 16 + n;
      v2f b;
      b.x = Wed[(size_t)k0 * 64 + col];
      b.y = Wed[(size_t)(k0 + 1) * 64 + col];
      g[t] = wmma4(a, b, g[t]);
    }
  }
#pragma unroll
  for (int t = 0; t < 4; ++t)
#pragma unroll
    for (int r = 0; r < 8; ++r)
      lds[(rbase + r) * LDS_ST + t * 16 + n] = g[t][r];
}

// --------------------- pass 1: alpha logits + atomic segment max over dst
__global__ void __launch_bounds__(128) k_edge_alpha(
    const float* __restrict__ attr, const int* __restrict__ ei,
    const float* __restrict__ We1, const float* __restrict__ be1,
    const float* __restrict__ We2, const float* __restrict__ be2,
    const float* __restrict__ Wed, const float* __restrict__ Q,
    const float* __restrict__ Kp, float* __restrict__ ALPHA,
    unsigned* __restrict__ AMAX) {
  __shared__ float smem[4 * 16 * LDS_ST];
  int wave = threadIdx.x >> 5;
  int lane = threadIdx.x & 31;
  float* lds = smem + wave * 16 * LDS_ST;
  int e0 = (blockIdx.x * 4 + wave) * 16;

  edge_tile_e(e0, lane, lds, attr, We1, be1, We2, be2, Wed);

  // each (lane,iter) handles one (edge, head): dot over C=16 feats, /sqrt(16)
#pragma unroll
  for (int it = 0; it < 2; ++it) {
    int m = lane & 15;
    int h = (lane >> 4) + 2 * it;
    int eid = e0 + m;
    int s = ei[eid];        // src
    int dd = ei[EE + eid];  // dst
    const float4* q4 = (const float4*)(Q + (size_t)dd * 64 + h * 16);
    const float4* k4 = (const float4*)(Kp + (size_t)s * 64 + h * 16);
    const float4* e4 = (const float4*)(lds + m * LDS_ST + h * 16);
    float acc = 0.f;
#pragma unroll
    for (int f = 0; f < 4; ++f) {
      float4 q = q4[f], k = k4[f], e = e4[f];
      acc += q.x * (k.x + e.x) + q.y * (k.y + e.y) + q.z * (k.z + e.z) +
             q.w * (k.w + e.w);
    }
    float alpha = acc * 0.25f;  // / sqrt(C=16)
    ALPHA[(size_t)eid * HH + h] = alpha;
    atomicMax(&AMAX[dd * HH + h], encf(alpha));
  }
}

// ------------------------------- pass 2: ex = exp(a - amax); denom += ex
__global__ void k_softmax_den(const int* __restrict__ ei,
                              float* __restrict__ ALPHA,
                              const unsigned* __restrict__ AMAX,
                              float* __restrict__ DENOM) {
  int i = blockIdx.x * blockDim.x + threadIdx.x;
  if (i >= EE * HH) return;
  int eid = i >> 2;
  int h = i & 3;
  int dd = ei[EE + eid];
  float mx = decf(AMAX[dd * HH + h]);
  float ex = expf(ALPHA[i] - mx);
  ALPHA[i] = ex;  // overwrite logits with exp values
  atomicAdd(&DENOM[dd * HH + h], ex);
}

// --------- pass 3: recompute e, msg = (v[src]+e)*attn, scatter-add to dst
__global__ void __launch_bounds__(128) k_edge_aggregate(
    const float* __restrict__ attr, const int* __restrict__ ei,
    const float* __restrict__ We1, const float* __restrict__ be1,
    const float* __restrict__ We2, const float* __restrict__ be2,
    const float* __restrict__ Wed, const float* __restrict__ Vp,
    const float* __restrict__ ALPHA, const float* __restrict__ DENOM,
    float* __restrict__ OUT_ACC) {
  __shared__ float smem[4 * 16 * LDS_ST];
  int wave = threadIdx.x >> 5;
  int lane = threadIdx.x & 31;
  float* lds = smem + wave * 16 * LDS_ST;
  int e0 = (blockIdx.x * 4 + wave) * 16;

  edge_tile_e(e0, lane, lds, attr, We1, be1, We2, be2, Wed);

  for (int idx = lane; idx < 16 * 64; idx += 32) {
    int m = idx >> 6;  // edge in tile
    int f = idx & 63;  // feature
    int h = f >> 4;    // head
    int eid = e0 + m;
    int s = ei[eid];
    int dd = ei[EE + eid];
    float ex = ALPHA[(size_t)eid * HH + h];
    float den = DENOM[dd * HH + h] + 1e-16f;
    float attn = ex / den;
    float val = (Vp[(size_t)s * 64 + f] + lds[m * LDS_ST + f]) * attn;
    atomicAdd(&OUT_ACC[(size_t)dd * 64 + f], val);
  }
}

// ------------------------------------------- beta gating + final blend
__global__ void k_beta(const float* __restrict__ OUT_ACC,
                       const float* __restrict__ XR,
                       const float* __restrict__ Wbeta,
                       float* __restrict__ out) {
  int nidx = blockIdx.x * blockDim.x + threadIdx.x;
  if (nidx >= NN) return;
  float z = 0.f;
  for (int f = 0; f < 64; ++f) {
    float ov = OUT_ACC[(size_t)nidx * 64 + f];
    float xr = XR[(size_t)nidx * 64 + f];
    z += Wbeta[f] * ov + Wbeta[64 + f] * xr + Wbeta[128 + f] * (ov - xr);
  }
  float beta = 1.f / (1.f + expf(-z));
  for (int f = 0; f < 64; ++f) {
    float ov = OUT_ACC[(size_t)nidx * 64 + f];
    float xr = XR[(size_t)nidx * 64 + f];
    out[(size_t)nidx * 64 + f] = beta * xr + (1.f - beta) * ov;
  }
}

extern "C" void kernel_launch(void* const* d_in, const int* in_sizes, int n_in,
                              void* d_out, int out_size, void* d_ws,
                              size_t ws_size, hipStream_t stream) {
  const float* x = (const float*)d_in[0];
  const int* ei = (const int*)d_in[1];  // [2,E] int32 (JAX x64 disabled)
  const float* attr = (const float*)d_in[2];
  const float* We1 = (const float*)d_in[3];
  const float* be1 = (const float*)d_in[4];
  const float* We2 = (const float*)d_in[5];
  const float* be2 = (const float*)d_in[6];
  const float* Wq = (const float*)d_in[7];
  const float* bq = (const float*)d_in[8];
  const float* Wk = (const float*)d_in[9];
  const float* bk = (const float*)d_in[10];
  const float* Wv = (const float*)d_in[11];
  const float* bv = (const float*)d_in[12];
  const float* Wed = (const float*)d_in[13];
  const float* Wsk = (const float*)d_in[14];
  const float* bsk = (const float*)d_in[15];
  const float* Wbeta = (const float*)d_in[16];
  float* out = (float*)d_out;

  // workspace layout (floats)
  float* ws = (float*)d_ws;
  float* Q = ws;                          // N*64
  float* Kp = Q + (size_t)NN * 64;        // N*64
  float* Vp = Kp + (size_t)NN * 64;       // N*64
  float* XR = Vp + (size_t)NN * 64;       // N*64
  float* OUT_ACC = XR + (size_t)NN * 64;  // N*64
  float* ALPHA = OUT_ACC + (size_t)NN * 64;        // E*H
  unsigned* AMAX = (unsigned*)(ALPHA + (size_t)EE * HH);  // N*H
  float* DENOM = (float*)(AMAX + (size_t)NN * HH);        // N*H

  k_init<<<(NN * 64 + 255) / 256, 256, 0, stream>>>(OUT_ACC, AMAX, DENOM);

  dim3 g1(NN / 16);  // 50000 % 16 == 0
  k_nodegemm<<<g1, 32, 0, stream>>>(x, Wq, bq, Q);
  k_nodegemm<<<g1, 32, 0, stream>>>(x, Wk, bk, Kp);
  k_nodegemm<<<g1, 32, 0, stream>>>(x, Wv, bv, Vp);
  k_nodegemm<<<g1, 32, 0, stream>>>(x, Wsk, bsk, XR);

  k_edge_alpha<<<EE / 64, 128, 0, stream>>>(attr, ei, We1, be1, We2, be2, Wed,
                                            Q, Kp, ALPHA, AMAX);
  k_softmax_den<<<(EE * HH + 255) / 256, 256, 0, stream>>>(ei, ALPHA, AMAX,
                                                           DENOM);
  k_edge_aggregate<<<EE / 64, 128, 0, stream>>>(attr, ei, We1, be1, We2, be2,
                                                Wed, Vp, ALPHA, DENOM, OUT_ACC);
  k_beta<<<(NN + 255) / 256, 256, 0, stream>>>(OUT_ACC, XR, Wbeta, out);
}